// MPNN_58394375356591
// MI455X (gfx1250) — compile-verified
//
#include <hip/hip_runtime.h>
#include <math.h>

// ---------------------------------------------------------------------------
// MPNN (TAGConv x3 + Set2Set readout) for MI455X / gfx1250.
// Dense GEMMs use V_WMMA_F32_16X16X4_F32 (exact f32 matrix math - workload is
// memory bound at ~20 GFLOP / ~300MB, so low-precision WMMA buys nothing).
// Each GEMM block computes a 64x64 output tile (4 waves x 4 N-tiles each) to
// amortize weight staging; next-chunk A rows are prefetched (global_prefetch).
// Sparse propagation uses global_atomic_add_f32 scatter (node features fit in
// the 192MB L2, so gathers/atomics stay on-chip).
// ---------------------------------------------------------------------------

typedef float v2f __attribute__((ext_vector_type(2)));
typedef float v8f __attribute__((ext_vector_type(8)));

#define TK 128   // K-chunk staged in LDS per GEMM block

// ---------------- generic WMMA f32 GEMM ------------------------------------
// out[M x Nout](ldc) = act( in[M x K](lda) @ W[Nout x K]^T + bias (+ out) )
// act: 0=none, 1=relu, 2=prelu(a = prelu_a[0])
// block = 128 threads (4 waves). Block tile = 64(M) x 64(N).
// wave w owns rows m0+16w..m0+16w+15 and sweeps the four 16x16 N-tiles.
__global__ void gemm_wmma_f32(const float* __restrict__ in, int lda,
                              const float* __restrict__ W,
                              const float* __restrict__ bias,
                              float* __restrict__ out, int ldc,
                              int M, int K, int Nout,
                              int act, const float* __restrict__ prelu_a,
                              int accum) {
  __shared__ float As[64 * TK];
  __shared__ float Bs[64 * TK];
  const int tid  = threadIdx.x;
  const int wave = tid >> 5;
  const int lane = tid & 31;
  const int half = lane >> 4;   // K-pair select (A/B), +8 row select (C/D)
  const int l16  = lane & 15;   // M index (A) / N index (B, C/D)
  const int m0 = blockIdx.x * 64;
  const int n0 = blockIdx.y * 64;

  v8f acc[4] = {};
  for (int kc = 0; kc < K; kc += TK) {
    // stage A tile 64 x TK (zero-padded)
    for (int i = tid; i < 64 * TK; i += 128) {
      int r = i / TK, c = i - r * TK;
      int gr = m0 + r, gc = kc + c;
      As[i] = (gr < M && gc < K) ? in[(long)gr * lda + gc] : 0.0f;
    }
    // stage B tile: 64 rows of W (n-major, k-minor), zero-padded
    for (int i = tid; i < 64 * TK; i += 128) {
      int r = i / TK, c = i - r * TK;
      int gn = n0 + r, gc = kc + c;
      Bs[i] = (gn < Nout && gc < K) ? W[(long)gn * K + gc] : 0.0f;
    }
    // prefetch next K-chunk of the A stream (hides HBM latency behind WMMA)
    if (kc + TK < K) {
      for (int i = tid; i < 64 * 4; i += 128) {
        int r = i >> 2, c = (i & 3) * 32;
        int gr = m0 + r, gc = kc + TK + c;
        if (gr < M && gc < K) __builtin_prefetch(&in[(long)gr * lda + gc], 0, 1);
      }
    }
    __syncthreads();
    const float* as = &As[(wave * 16 + l16) * TK];
#pragma unroll 4
    for (int kk = 0; kk < TK; kk += 4) {
      v2f a;
      a.x = as[kk + 2 * half];      // A[m][k], k = kk + 2*half
      a.y = as[kk + 2 * half + 1];  // A[m][k+1]
#pragma unroll
      for (int n = 0; n < 4; ++n) {
        const float* bs = &Bs[(n * 16 + l16) * TK];
        v2f b;
        b.x = bs[kk + 2 * half];    // B[k][n=l16] = W[n][k]
        b.y = bs[kk + 2 * half + 1];
        acc[n] = __builtin_amdgcn_wmma_f32_16x16x4_f32(
            /*neg_a=*/false, a, /*neg_b=*/false, b,
            /*c_mod=*/(short)0, acc[n], /*reuse_a=*/false, /*reuse_b=*/false);
      }
    }
    __syncthreads();
  }

  // epilogue: C/D layout -> VGPR r holds M = r + 8*half, N = l16 per lane
  const float aslope = (act == 2) ? prelu_a[0] : 0.0f;
#pragma unroll
  for (int n = 0; n < 4; ++n) {
    const int col = n0 + n * 16 + l16;
    if (col < Nout) {
      const float bcol = bias ? bias[col] : 0.0f;
#pragma unroll
      for (int r = 0; r < 8; ++r) {
        int row = m0 + wave * 16 + r + 8 * half;
        if (row < M) {
          float v = acc[n][r] + bcol;
          if (accum) v += out[(long)row * ldc + col];
          if (act == 1)      v = v > 0.0f ? v : 0.0f;
          else if (act == 2) v = v >= 0.0f ? v : aslope * v;
          out[(long)row * ldc + col] = v;
        }
      }
    }
  }
}

// ---------------- graph / elementwise kernels ------------------------------
__global__ void fill_kernel(float* __restrict__ p, float v, long n) {
  long i = (long)blockIdx.x * blockDim.x + threadIdx.x;
  if (i < n) p[i] = v;
}

__global__ void fill2d_kernel(float* __restrict__ p, int ld, int rows, int cols, float v) {
  long t = (long)blockIdx.x * blockDim.x + threadIdx.x;
  if (t >= (long)rows * cols) return;
  int r = (int)(t / cols), c = (int)(t - (long)r * cols);
  p[(long)r * ld + c] = v;
}

__global__ void copy2d_kernel(const float* __restrict__ src, int sld,
                              float* __restrict__ dst, int dld, int rows, int cols) {
  long t = (long)blockIdx.x * blockDim.x + threadIdx.x;
  if (t >= (long)rows * cols) return;
  int r = (int)(t / cols), c = (int)(t - (long)r * cols);
  dst[(long)r * dld + c] = src[(long)r * sld + c];
}

__global__ void deg_kernel(const int* __restrict__ src, const int* __restrict__ dst,
                           float* __restrict__ deg_out, float* __restrict__ deg_in,
                           int E, float w) {
  int e = blockIdx.x * blockDim.x + threadIdx.x;
  if (e >= E) return;
  atomicAdd(&deg_out[src[e]], w);
  atomicAdd(&deg_in[dst[e]], w);
}

__global__ void wn_kernel(const int* __restrict__ src, const int* __restrict__ dst,
                          const float* __restrict__ deg_out, const float* __restrict__ deg_in,
                          float* __restrict__ wn, int E, float w) {
  int e = blockIdx.x * blockDim.x + threadIdx.x;
  if (e >= E) return;
  wn[e] = w * rsqrtf(fmaxf(deg_out[src[e]], 1e-9f))
            * rsqrtf(fmaxf(deg_in[dst[e]], 1e-9f));
}

// one wave per edge, 2 feature columns per lane (HID=64)
__global__ void prop_kernel(const float* __restrict__ hin, int in_ld,
                            float* __restrict__ hout, int out_ld,
                            const int* __restrict__ src, const int* __restrict__ dst,
                            const float* __restrict__ wn, int E) {
  long t = (long)blockIdx.x * blockDim.x + threadIdx.x;
  int e = (int)(t >> 5);
  int lane = (int)(t & 31);
  if (e >= E) return;
  const int s = src[e], d = dst[e];
  const float w = wn[e];
  const float x0 = hin[(long)s * in_ld + lane];
  const float x1 = hin[(long)s * in_ld + lane + 32];
  atomicAdd(&hout[(long)d * out_ld + lane],      x0 * w);
  atomicAdd(&hout[(long)d * out_ld + lane + 32], x1 * w);
}

// ---------------- Set2Set ---------------------------------------------------
__device__ __forceinline__ float sigmoidf_(float x) { return 1.0f / (1.0f + expf(-x)); }

// order-preserving float<->uint key for atomicMax on floats
__device__ __forceinline__ unsigned int fkey(float f) {
  unsigned int u = __float_as_uint(f);
  return (u & 0x80000000u) ? ~u : (u | 0x80000000u);
}
__device__ __forceinline__ float fdecode(unsigned int k) {
  unsigned int u = (k & 0x80000000u) ? (k ^ 0x80000000u) : ~k;
  return __uint_as_float(u);
}

// gates[B,512] (i,f,g,o) -> update cx,hx; write q into q_star[:,0:128]
__global__ void lstm_kernel(const float* __restrict__ gates,
                            float* __restrict__ hx, float* __restrict__ cx,
                            float* __restrict__ qstar, int B) {
  long t = (long)blockIdx.x * blockDim.x + threadIdx.x;
  if (t >= (long)B * 128) return;
  int b = (int)(t >> 7), j = (int)(t & 127);
  const float* gr = gates + (long)b * 512;
  float gi = gr[j], gf = gr[128 + j], gg = gr[256 + j], go = gr[384 + j];
  float c = sigmoidf_(gf) * cx[t] + sigmoidf_(gi) * tanhf(gg);
  cx[t] = c;
  float h = sigmoidf_(go) * tanhf(c);
  hx[t] = h;
  qstar[(long)b * 256 + j] = h;
}

// one wave per node: sc[n] = <feat[n,:], hx[g,:]> (d=128), then segment max
__global__ void score_kernel(const float* __restrict__ feat, const float* __restrict__ hx,
                             const int* __restrict__ gids, float* __restrict__ sc,
                             unsigned int* __restrict__ gmaxkey, int N) {
  long t = (long)blockIdx.x * blockDim.x + threadIdx.x;
  int n = (int)(t >> 5), lane = (int)(t & 31);
  if (n >= N) return;
  const int g = gids[n];
  const float* f = feat + (long)n * 128;
  const float* q = hx + (long)g * 128;
  float v = f[lane] * q[lane] + f[lane + 32] * q[lane + 32]
          + f[lane + 64] * q[lane + 64] + f[lane + 96] * q[lane + 96];
#pragma unroll
  for (int off = 16; off > 0; off >>= 1) v += __shfl_xor(v, off, 32);
  if (lane == 0) {
    sc[n] = v;
    atomicMax(&gmaxkey[g], fkey(v));
  }
}

__global__ void softmax_exp_kernel(const int* __restrict__ gids, float* __restrict__ sc,
                                   const unsigned int* __restrict__ gmaxkey,
                                   float* __restrict__ gsum, int N) {
  int n = blockIdx.x * blockDim.x + threadIdx.x;
  if (n >= N) return;
  int g = gids[n];
  float ex = expf(sc[n] - fdecode(gmaxkey[g]));
  sc[n] = ex;
  atomicAdd(&gsum[g], ex);
}

// r = segment_sum(feat * alpha) accumulated into q_star[:,128:256]
__global__ void readout_kernel(const float* __restrict__ feat, const int* __restrict__ gids,
                               const float* __restrict__ sc, const float* __restrict__ gsum,
                               float* __restrict__ qstar, int N) {
  long t = (long)blockIdx.x * blockDim.x + threadIdx.x;
  int n = (int)(t >> 5), lane = (int)(t & 31);
  if (n >= N) return;
  const int g = gids[n];
  const float alpha = sc[n] / gsum[g];
  const float* f = feat + (long)n * 128;
  float* r = qstar + (long)g * 256 + 128;
  atomicAdd(&r[lane],      f[lane]      * alpha);
  atomicAdd(&r[lane + 32], f[lane + 32] * alpha);
  atomicAdd(&r[lane + 64], f[lane + 64] * alpha);
  atomicAdd(&r[lane + 96], f[lane + 96] * alpha);
}

// ---------------------------------------------------------------------------
static inline int cdiv(long a, long b) { return (int)((a + b - 1) / b); }

extern "C" void kernel_launch(void* const* d_in, const int* in_sizes, int n_in,
                              void* d_out, int out_size, void* d_ws, size_t ws_size,
                              hipStream_t stream) {
  (void)n_in; (void)ws_size;
  const float* node_attr = (const float*)d_in[0];
  const int*   src       = (const int*)d_in[1];
  const int*   dst       = (const int*)d_in[2];
  const int*   gids      = (const int*)d_in[3];
  const float* W_proj = (const float*)d_in[5];
  const float* b_proj = (const float*)d_in[6];
  const float* W_tag  = (const float*)d_in[7];
  const float* b_tag  = (const float*)d_in[8];
  const float* W_gru1 = (const float*)d_in[9];
  const float* b_gru1 = (const float*)d_in[10];
  const float* W_gru2 = (const float*)d_in[11];
  const float* b_gru2 = (const float*)d_in[12];
  const float* Wih    = (const float*)d_in[13];
  const float* bih    = (const float*)d_in[14];
  const float* Whh    = (const float*)d_in[15];
  const float* bhh    = (const float*)d_in[16];
  const float* W_sp   = (const float*)d_in[17];
  const float* b_sp   = (const float*)d_in[18];
  const float* prelu_a= (const float*)d_in[19];

  const int NODE_IN = 74, HID = 64;
  const int N = in_sizes[0] / NODE_IN;
  const int E = in_sizes[1];
  const int G = out_size / 128;          // RD = 128
  const float SIG1 = 0.73105857863000490f;   // sigmoid(1.0)

  // ---- workspace carve-up (floats) ----
  float* ws = (float*)d_ws;
  long off = 0;
  float* wn     = ws + off; off += E;
  float* degO   = ws + off; off += N;
  float* degI   = ws + off; off += N;
  float* feat   = ws + off; off += (long)N * 128;   // [h0 | nf]
  float* hid    = ws + off; off += (long)N * 64;
  float* nf     = ws + off; off += (long)N * 64;
  float* tcat   = ws + off; off += (long)N * 192;   // [h | Ah | A2h]
  float* cat    = ws + off; off += (long)N * 128;   // [t | hid]
  float* qstar  = ws + off; off += (long)G * 256;   // [q | r]
  float* hx     = ws + off; off += (long)G * 128;
  float* cx     = ws + off; off += (long)G * 128;
  float* gates  = ws + off; off += (long)G * 512;
  float* sc     = ws + off; off += N;
  float* gmaxk  = ws + off; off += G;               // used as uint keys
  float* gsum   = ws + off; off += G;

  const int B256 = 256;
  dim3 gblk(128);

  // ---- edge weight normalization ----
  fill_kernel<<<cdiv(2L * N, B256), B256, 0, stream>>>(degO, 0.0f, 2L * N);
  deg_kernel<<<cdiv(E, B256), B256, 0, stream>>>(src, dst, degO, degI, E, SIG1);
  wn_kernel<<<cdiv(E, B256), B256, 0, stream>>>(src, dst, degO, degI, wn, E, SIG1);

  // ---- h0 = relu(node_attr @ W_proj^T + b) -> feat[:,0:64] ----
  gemm_wmma_f32<<<dim3(cdiv(N, 64), 1), gblk, 0, stream>>>(
      node_attr, NODE_IN, W_proj, b_proj, feat, 128, N, NODE_IN, HID, 1, nullptr, 0);
  copy2d_kernel<<<cdiv((long)N * 64, B256), B256, 0, stream>>>(feat, 128, nf, 64, N, 64);
  copy2d_kernel<<<cdiv((long)N * 64, B256), B256, 0, stream>>>(feat, 128, hid, 64, N, 64);

  // ---- 3 message-passing layers ----
  for (int it = 0; it < 3; ++it) {
    fill_kernel<<<cdiv((long)N * 192, B256), B256, 0, stream>>>(tcat, 0.0f, (long)N * 192);
    copy2d_kernel<<<cdiv((long)N * 64, B256), B256, 0, stream>>>(nf, 64, tcat, 192, N, 64);
    prop_kernel<<<cdiv((long)E * 32, B256), B256, 0, stream>>>(tcat, 192, tcat + 64, 192, src, dst, wn, E);
    prop_kernel<<<cdiv((long)E * 32, B256), B256, 0, stream>>>(tcat + 64, 192, tcat + 128, 192, src, dst, wn, E);
    // t = relu(lin([h,Ah,A2h])) -> cat[:,0:64]
    gemm_wmma_f32<<<dim3(cdiv(N, 64), 1), gblk, 0, stream>>>(
        tcat, 192, W_tag, b_tag, cat, 128, N, 192, HID, 1, nullptr, 0);
    copy2d_kernel<<<cdiv((long)N * 64, B256), B256, 0, stream>>>(hid, 64, cat + 64, 128, N, 64);
    gemm_wmma_f32<<<dim3(cdiv(N, 64), 1), gblk, 0, stream>>>(
        cat, 128, W_gru1, b_gru1, nf, 64, N, 128, HID, 0, nullptr, 0);
    gemm_wmma_f32<<<dim3(cdiv(N, 64), 1), gblk, 0, stream>>>(
        cat, 128, W_gru2, b_gru2, hid, 64, N, 128, HID, 0, nullptr, 0);
  }
  // feat = [h0 | nf]
  copy2d_kernel<<<cdiv((long)N * 64, B256), B256, 0, stream>>>(nf, 64, feat + 64, 128, N, 64);

  // ---- Set2Set readout (3 iterations) ----
  fill_kernel<<<cdiv((long)G * 512, B256), B256, 0, stream>>>(qstar, 0.0f, (long)G * 512); // qstar,hx,cx
  for (int it = 0; it < 3; ++it) {
    gemm_wmma_f32<<<dim3(cdiv(G, 64), 8), gblk, 0, stream>>>(
        qstar, 256, Wih, bih, gates, 512, G, 256, 512, 0, nullptr, 0);
    gemm_wmma_f32<<<dim3(cdiv(G, 64), 8), gblk, 0, stream>>>(
        hx, 128, Whh, bhh, gates, 512, G, 128, 512, 0, nullptr, 1);
    lstm_kernel<<<cdiv((long)G * 128, B256), B256, 0, stream>>>(gates, hx, cx, qstar, G);
    fill_kernel<<<cdiv(2L * G, B256), B256, 0, stream>>>(gmaxk, 0.0f, 2L * G); // keys(=min) + gsum
    fill2d_kernel<<<cdiv((long)G * 128, B256), B256, 0, stream>>>(qstar + 128, 256, G, 128, 0.0f);
    score_kernel<<<cdiv((long)N * 32, B256), B256, 0, stream>>>(
        feat, hx, gids, sc, (unsigned int*)gmaxk, N);
    softmax_exp_kernel<<<cdiv(N, B256), B256, 0, stream>>>(
        gids, sc, (const unsigned int*)gmaxk, gsum, N);
    readout_kernel<<<cdiv((long)N * 32, B256), B256, 0, stream>>>(feat, gids, sc, gsum, qstar, N);
  }

  // ---- out = PReLU(q_star @ W_sp^T + b_sp) ----
  gemm_wmma_f32<<<dim3(cdiv(G, 64), 2), gblk, 0, stream>>>(
      qstar, 256, W_sp, b_sp, (float*)d_out, 128, G, 256, 128, 2, prelu_a, 0);
}